// PointNet2Down_67997922230566
// MI455X (gfx1250) — compile-verified
//
#include <hip/hip_runtime.h>

typedef float v2f __attribute__((ext_vector_type(2)));
typedef float v8f __attribute__((ext_vector_type(8)));

#define NB     4
#define NPTS   8192
#define NCTR   2048
#define NSAMP  32
#define RAD2   0.04f
#define CFEAT  64
#define COUT   67   // 3 + 64

// ---------------------------------------------------------------------------
// Kernel 1: farthest point sampling. One workgroup per batch, 1024 threads.
// All point state (coords + running min-d2) lives in VGPRs: 8 points/thread.
// The argmax butterfly carries (val, idx, x, y, z) so the next centroid's
// coordinates fall out of the reduction — no big LDS arrays at all.
// ---------------------------------------------------------------------------
__global__ __launch_bounds__(1024) void fps_pn2_kernel(
    const float* __restrict__ xyz, float* __restrict__ centers)
{
    __shared__ float rv[32], rx[32], ry[32], rz[32];
    __shared__ int   ri[32];
    __shared__ float selx, sely, selz;

    const int b    = blockIdx.x;
    const int tid  = threadIdx.x;
    const int lane = tid & 31;
    const int wave = tid >> 5;
    const float* xb = xyz + (size_t)b * NPTS * 3;

    // thread t owns points i = k*1024 + t  (coalesced loads)
    float X[8], Y[8], Z[8], M[8];
    #pragma unroll
    for (int k = 0; k < 8; ++k) {
        const int i = k * 1024 + tid;
        X[k] = xb[i * 3 + 0];
        Y[k] = xb[i * 3 + 1];
        Z[k] = xb[i * 3 + 2];
        M[k] = 1e10f;
    }
    if (tid == 0) { selx = X[0]; sely = Y[0]; selz = Z[0]; }  // farthest=0
    __syncthreads();

    for (int s = 0; s < NCTR; ++s) {
        const float cx = selx, cy = sely, cz = selz;
        if (tid == 0) {
            float* c = centers + ((size_t)b * NCTR + s) * 4;
            c[0] = cx; c[1] = cy; c[2] = cz;
            c[3] = cx * cx + cy * cy + cz * cz;
        }
        float best = -1.0f, bx = 0.f, by = 0.f, bz = 0.f;
        int   bi   = 0;
        #pragma unroll
        for (int k = 0; k < 8; ++k) {
            const float dx = X[k] - cx, dy = Y[k] - cy, dz = Z[k] - cz;
            const float d2 = dx * dx + dy * dy + dz * dz;
            const float m  = fminf(M[k], d2);
            M[k] = m;
            if (m > best) {
                best = m; bi = k * 1024 + tid;
                bx = X[k]; by = Y[k]; bz = Z[k];
            }
        }
        // wave32 butterfly argmax carrying coords (ties -> smaller index)
        #pragma unroll
        for (int off = 16; off >= 1; off >>= 1) {
            const float ov = __shfl_xor(best, off, 32);
            const int   oi = __shfl_xor(bi,   off, 32);
            const float ox = __shfl_xor(bx,   off, 32);
            const float oy = __shfl_xor(by,   off, 32);
            const float oz = __shfl_xor(bz,   off, 32);
            if (ov > best || (ov == best && oi < bi)) {
                best = ov; bi = oi; bx = ox; by = oy; bz = oz;
            }
        }
        if (lane == 0) {
            rv[wave] = best; ri[wave] = bi;
            rx[wave] = bx;   ry[wave] = by; rz[wave] = bz;
        }
        __syncthreads();
        if (wave == 0) {
            float v  = rv[lane], vx = rx[lane], vy = ry[lane], vz = rz[lane];
            int   i2 = ri[lane];
            #pragma unroll
            for (int off = 16; off >= 1; off >>= 1) {
                const float ov = __shfl_xor(v,  off, 32);
                const int   oi = __shfl_xor(i2, off, 32);
                const float ox = __shfl_xor(vx, off, 32);
                const float oy = __shfl_xor(vy, off, 32);
                const float oz = __shfl_xor(vz, off, 32);
                if (ov > v || (ov == v && oi < i2)) {
                    v = ov; i2 = oi; vx = ox; vy = oy; vz = oz;
                }
            }
            if (lane == 0) { selx = vx; sely = vy; selz = vz; }
        }
        __syncthreads();
    }
}

// ---------------------------------------------------------------------------
// Ballot mux: deliver row m's 16-bit hit mask to lane m.
// Row m (m<8): low half of ballot[m]; row m (>=8): high half of ballot[m-8].
// ---------------------------------------------------------------------------
__device__ __forceinline__ unsigned row_mask16(
    int m, unsigned b0, unsigned b1, unsigned b2, unsigned b3,
    unsigned b4, unsigned b5, unsigned b6, unsigned b7)
{
    unsigned s01 = (m & 1) ? b1 : b0;
    unsigned s23 = (m & 1) ? b3 : b2;
    unsigned s45 = (m & 1) ? b5 : b4;
    unsigned s67 = (m & 1) ? b7 : b6;
    unsigned sA  = (m & 2) ? s23 : s01;
    unsigned sB  = (m & 2) ? s67 : s45;
    unsigned bal = (m & 4) ? sB : sA;
    return (m & 8) ? (bal >> 16) : (bal & 0xFFFFu);
}

// ---------------------------------------------------------------------------
// Kernel 2: ball query via V_WMMA_F32_16X16X4_F32 + fused gather/concat.
// 256 threads (8 waves); each wave owns 16 centers. Points staged in LDS as
// (-2x,-2y,-2z,|p|^2). Two independent 16-pt tiles per iteration: both B
// loads issue, two back-to-back WMMAs, one 32-bit per-row mask, ballot-based
// early exit once all 16 rows hold 32 samples.
// ---------------------------------------------------------------------------
__global__ __launch_bounds__(256) void ballq_pn2_kernel(
    const float* __restrict__ xyz, const float* __restrict__ feat,
    const float* __restrict__ centers, float* __restrict__ out)
{
    extern __shared__ float sm2[];
    float* pts    = sm2;                       // NPTS * 4 floats (131072 B)
    int*   rowbuf = (int*)(sm2 + NPTS * 4);    // 8 waves * 16 rows * 32 ints

    const int b     = blockIdx.x >> 4;   // 16 blocks per batch
    const int tileS = blockIdx.x & 15;   // group of 128 centers
    const int tid   = threadIdx.x;
    const int lane  = tid & 31;
    const int wave  = tid >> 5;

    const float* xb = xyz + (size_t)b * NPTS * 3;
    for (int i = tid; i < NPTS; i += 256) {
        float x = xb[i * 3 + 0], y = xb[i * 3 + 1], z = xb[i * 3 + 2];
        ((float4*)pts)[i] = make_float4(-2.f * x, -2.f * y, -2.f * z,
                                        x * x + y * y + z * z);
    }
    __syncthreads();

    const int  s0 = tileS * 128 + wave * 16;
    const int  m  = lane & 15;
    const bool hi = lane >= 16;
    const float* cb = centers + ((size_t)b * NCTR + s0) * 4;

    // A (16x4): lane m holds (K0,K1)=(cx,cy) low half; (K2,K3)=(cz,1) high
    v2f a;
    {
        const float* c = cb + m * 4;
        a[0] = hi ? c[2] : c[0];
        a[1] = hi ? 1.0f : c[1];
    }
    // C: VGPR i = row i (lanes 0-15) / row i+8 (lanes 16-31): |c|^2
    v8f cacc;
    #pragma unroll
    for (int i = 0; i < 8; ++i)
        cacc[i] = cb[(i + (hi ? 8 : 0)) * 4 + 3];

    int  cnt   = hi ? NSAMP : 0;   // upper lanes always "done"
    int* myrow = rowbuf + (wave * 16 + m) * NSAMP;

    for (int t = 0; t < NPTS / 32; ++t) {
        const int n0 = (t << 5) | m;          // tile 0 point
        const float* pp0 = pts + n0 * 4 + (hi ? 2 : 0);
        const float* pp1 = pp0 + 64;          // tile 1: +16 points
        v2f b0; b0[0] = pp0[0]; b0[1] = pp0[1];
        v2f b1; b1[0] = pp1[0]; b1[1] = pp1[1];

        v8f d0 = __builtin_amdgcn_wmma_f32_16x16x4_f32(
            false, a, false, b0, (short)0, cacc, false, false);
        v8f d1 = __builtin_amdgcn_wmma_f32_16x16x4_f32(
            false, a, false, b1, (short)0, cacc, false, false);

        unsigned lo = row_mask16(m,
            __builtin_amdgcn_ballot_w32(d0[0] <= RAD2),
            __builtin_amdgcn_ballot_w32(d0[1] <= RAD2),
            __builtin_amdgcn_ballot_w32(d0[2] <= RAD2),
            __builtin_amdgcn_ballot_w32(d0[3] <= RAD2),
            __builtin_amdgcn_ballot_w32(d0[4] <= RAD2),
            __builtin_amdgcn_ballot_w32(d0[5] <= RAD2),
            __builtin_amdgcn_ballot_w32(d0[6] <= RAD2),
            __builtin_amdgcn_ballot_w32(d0[7] <= RAD2));
        unsigned hi16 = row_mask16(m,
            __builtin_amdgcn_ballot_w32(d1[0] <= RAD2),
            __builtin_amdgcn_ballot_w32(d1[1] <= RAD2),
            __builtin_amdgcn_ballot_w32(d1[2] <= RAD2),
            __builtin_amdgcn_ballot_w32(d1[3] <= RAD2),
            __builtin_amdgcn_ballot_w32(d1[4] <= RAD2),
            __builtin_amdgcn_ballot_w32(d1[5] <= RAD2),
            __builtin_amdgcn_ballot_w32(d1[6] <= RAD2),
            __builtin_amdgcn_ballot_w32(d1[7] <= RAD2));
        unsigned mask = lo | (hi16 << 16);

        const int base = t << 5;
        while (mask && cnt < NSAMP) {
            const int bpos = __builtin_ctz(mask);
            mask &= mask - 1;
            myrow[cnt++] = base + bpos;
        }
        if (__builtin_amdgcn_ballot_w32(cnt >= NSAMP) == 0xFFFFFFFFu) break;
    }

    // pad with first valid index (0 if none) — matches reference semantics
    if (!hi) {
        int first = (cnt > 0) ? myrow[0] : 0;
        for (int k = cnt; k < NSAMP; ++k) myrow[k] = first;
    }
    __syncthreads();

    // fused gather + concat: per row, 32 lanes handle one sample each
    for (int r = 0; r < 16; ++r) {
        const int s = s0 + r;
        const float* c = centers + ((size_t)b * NCTR + s) * 4;
        const float ccx = c[0], ccy = c[1], ccz = c[2];
        const int idx = rowbuf[(wave * 16 + r) * NSAMP + lane];
        const float* pv = xyz + ((size_t)b * NPTS + idx) * 3;
        float* o = out + (((size_t)b * NCTR + s) * NSAMP + lane) * COUT;
        o[0] = pv[0] - ccx;
        o[1] = pv[1] - ccy;
        o[2] = pv[2] - ccz;
        const float* f = feat + ((size_t)b * NPTS + idx) * CFEAT; // 256B aligned
        #pragma unroll
        for (int cch = 0; cch < CFEAT; cch += 4) {
            float4 v = *(const float4*)(f + cch);
            o[3 + cch + 0] = v.x;
            o[3 + cch + 1] = v.y;
            o[3 + cch + 2] = v.z;
            o[3 + cch + 3] = v.w;
        }
    }
}

extern "C" void kernel_launch(void* const* d_in, const int* in_sizes, int n_in,
                              void* d_out, int out_size, void* d_ws, size_t ws_size,
                              hipStream_t stream) {
    (void)in_sizes; (void)n_in; (void)out_size; (void)ws_size;
    const float* xyz      = (const float*)d_in[0];
    const float* features = (const float*)d_in[1];
    float* out     = (float*)d_out;
    float* centers = (float*)d_ws;   // [4][2048][4] = 128 KB

    fps_pn2_kernel<<<NB, 1024, 0, stream>>>(xyz, centers);

    const size_t bqLds = (size_t)NPTS * 4 * sizeof(float)
                       + (size_t)8 * 16 * NSAMP * sizeof(int);   // 144 KB
    ballq_pn2_kernel<<<NB * 16, 256, bqLds, stream>>>(xyz, features, centers, out);
}